// OuterProductMHSA_85469849191014
// MI455X (gfx1250) — compile-verified
//
#include <hip/hip_runtime.h>

// Problem constants from the reference: B=4, N=1024, D=128.
#define BATCH 4
#define NSEQ  1024
#define DIM   128
#define ROWS_PER_BLOCK 128                            // rows of X per workgroup
#define ROWBLOCKS_PER_BATCH (NSEQ / ROWS_PER_BLOCK)   // 8
#define XS_STRIDE 132        // padded X row stride (floats): conflict-free A-frag b64s
#define WP_STRIDE 288        // K-pair-interleaved W row stride (floats): 2*DIM + 32
                             // -> half-lanes map to disjoint 32-bank sets for B-frag b64s

typedef float v2f __attribute__((ext_vector_type(2)));
typedef float v8f __attribute__((ext_vector_type(8)));
typedef int   b128v __attribute__((vector_size(4 * sizeof(int))));  // matches builtin param

#define GLOBAL_AS __attribute__((address_space(1)))
#define LDS_AS    __attribute__((address_space(3)))

// gfx1250 async global->LDS copy path (ASYNCcnt-tracked), with safe fallback.
#if defined(__has_builtin)
#if __has_builtin(__builtin_amdgcn_global_load_async_to_lds_b128) && \
    __has_builtin(__builtin_amdgcn_s_wait_asynccnt)
#define USE_ASYNC_LDS 1
#endif
#endif
#ifndef USE_ASYNC_LDS
#define USE_ASYNC_LDS 0
#endif

// Stage the 128x128 X tile into LDS (row stride XS_STRIDE).
__device__ __forceinline__ void stage_x_tile(const float* __restrict__ Xg,
                                             float* __restrict__ Xs, int t) {
#if USE_ASYNC_LDS
    for (int i = t; i < ROWS_PER_BLOCK * DIM / 4; i += 256) {
        const int row = i >> 5;          // DIM/4 = 32 float4 per row
        const int c4  = i & 31;
        __builtin_amdgcn_global_load_async_to_lds_b128(
            (GLOBAL_AS b128v*)(Xg + row * DIM + c4 * 4),
            (LDS_AS b128v*)(Xs + row * XS_STRIDE + c4 * 4),
            /*offset=*/0, /*cpol=*/0);
    }
#else
    for (int i = t; i < ROWS_PER_BLOCK * DIM / 4; i += 256) {
        const int row = i >> 5;
        const int c4  = i & 31;
        float4 vv = ((const float4*)Xg)[row * (DIM / 4) + c4];
        *(float4*)&Xs[row * XS_STRIDE + c4 * 4] = vv;
    }
#endif
}

__device__ __forceinline__ void stage_wait_and_barrier() {
#if USE_ASYNC_LDS
    __builtin_amdgcn_s_wait_asynccnt(0);
#endif
    __syncthreads();
}

// Stage W [DIM x DIM] row-major into K-pair-interleaved LDS layout:
//   Wp[j*WP_STRIDE + 2*n + e] = W[2*j + e][n],  j = k>>1, e = k&1
// so a B fragment {W[kb][col], W[kb+1][col]} (kb even) is one contiguous v2f.
__device__ __forceinline__ void stage_w_pairs(const float* __restrict__ W,
                                              float* __restrict__ Wp, int t) {
    for (int i = t; i < (DIM / 2) * (DIM / 2); i += 256) {   // 64*64 = 4096
        const int jp = i >> 6;     // k-pair row 0..63
        const int np = i & 63;     // n-pair   0..63
        const float2 g0 = *(const float2*)&W[(2 * jp    ) * DIM + 2 * np];
        const float2 g1 = *(const float2*)&W[(2 * jp + 1) * DIM + 2 * np];
        *(float2*)&Wp[jp * WP_STRIDE + 4 * np    ] = make_float2(g0.x, g1.x);
        *(float2*)&Wp[jp * WP_STRIDE + 4 * np + 2] = make_float2(g0.y, g1.y);
    }
}

// ---------------------------------------------------------------------------
// Pass 1: k = X@Wk + bk, v = X@Wv + bv via V_WMMA_F32_16X16X4_F32, reduced to
// per-block column partials of sum_rows(k*v). Each wave owns one 16-column
// slab and accumulates 4 row-tiles per pass so B fragments are loaded once
// per K-step for 8 WMMAs.
// ---------------------------------------------------------------------------
__global__ __launch_bounds__(256)
void kv_partial_kernel(const float* __restrict__ X,
                       const float* __restrict__ Wk, const float* __restrict__ bk,
                       const float* __restrict__ Wv, const float* __restrict__ bv,
                       float* __restrict__ partial)
{
    extern __shared__ float smem[];
    float* Xs  = smem;                                 // 128 x XS_STRIDE
    float* Wkp = Xs  + ROWS_PER_BLOCK * XS_STRIDE;     // 64 x WP_STRIDE
    float* Wvp = Wkp + (DIM / 2) * WP_STRIDE;          // 64 x WP_STRIDE

    const int b       = blockIdx.x >> 3;
    const int rb      = blockIdx.x & 7;
    const int rowBase = rb * ROWS_PER_BLOCK;
    const float* Xg   = X + ((size_t)b * NSEQ + rowBase) * DIM;

    const int t = threadIdx.x;
    stage_x_tile(Xg, Xs, t);
    stage_w_pairs(Wk, Wkp, t);
    stage_w_pairs(Wv, Wvp, t);
    stage_wait_and_barrier();

    const int wave = t >> 5;
    const int lane = t & 31;
    const int nl   = lane & 15;    // N within tile; also M row for A-frag
    const int half = lane >> 4;    // selects K pair per ISA layout
    const int col  = wave * 16 + nl;

    const float bkc = bk[col];
    const float bvc = bv[col];

    // Per-lane B-fragment bases: element (kpair = 2*kk + half, 2*col).
    const float* Bk = &Wkp[half * WP_STRIDE + 2 * col];
    const float* Bv = &Wvp[half * WP_STRIDE + 2 * col];

    const v8f vzero = {};
    float colsum = 0.0f;

    for (int mb = 0; mb < 2; ++mb) {       // two blocks of 4 row-tiles
        v8f ck[4], cv[4];
        const float* Arow[4];
        #pragma unroll
        for (int mm = 0; mm < 4; ++mm) {
            ck[mm] = vzero;
            cv[mm] = vzero;
            Arow[mm] = &Xs[((mb * 4 + mm) * 16 + nl) * XS_STRIDE + 2 * half];
        }

        #pragma unroll 4
        for (int kk = 0; kk < DIM / 4; ++kk) {
            v2f bfk = *(const v2f*)(Bk + (size_t)kk * 2 * WP_STRIDE);
            v2f bfv = *(const v2f*)(Bv + (size_t)kk * 2 * WP_STRIDE);
            #pragma unroll
            for (int mm = 0; mm < 4; ++mm) {
                v2f a = *(const v2f*)(Arow[mm] + kk * 4);
                ck[mm] = __builtin_amdgcn_wmma_f32_16x16x4_f32(false, a, false, bfk,
                                                               (short)0, ck[mm], false, false);
                cv[mm] = __builtin_amdgcn_wmma_f32_16x16x4_f32(false, a, false, bfv,
                                                               (short)0, cv[mm], false, false);
            }
        }

        #pragma unroll
        for (int mm = 0; mm < 4; ++mm) {
            float s = 0.0f;
            #pragma unroll
            for (int r = 0; r < 8; ++r)
                s += (ck[mm][r] + bkc) * (cv[mm][r] + bvc);
            s += __shfl_xor(s, 16);        // fold M=0..7 with M=8..15 (wave32)
            colsum += s;
        }
    }

    if (lane < 16)
        partial[(size_t)blockIdx.x * DIM + col] = colsum;
}

// ---------------------------------------------------------------------------
// Pass 2: deterministic reduction of the 8 row-block partials per batch.
// ---------------------------------------------------------------------------
__global__ __launch_bounds__(128)
void kv_reduce_kernel(const float* __restrict__ partial, float* __restrict__ kv)
{
    const int b = blockIdx.x;
    const int d = threadIdx.x;
    float s = 0.0f;
    #pragma unroll
    for (int rb = 0; rb < ROWBLOCKS_PER_BATCH; ++rb)
        s += partial[(size_t)(b * ROWBLOCKS_PER_BATCH + rb) * DIM + d];
    kv[b * DIM + d] = s;
}

// ---------------------------------------------------------------------------
// Pass 3: q = X@Wq + bq via WMMA, out = q * kv[b, col].
// ---------------------------------------------------------------------------
__global__ __launch_bounds__(256)
void q_out_kernel(const float* __restrict__ X,
                  const float* __restrict__ Wq, const float* __restrict__ bq,
                  const float* __restrict__ kv, float* __restrict__ out)
{
    extern __shared__ float smem[];
    float* Xs  = smem;
    float* Wqp = Xs + ROWS_PER_BLOCK * XS_STRIDE;

    const int b       = blockIdx.x >> 3;
    const int rb      = blockIdx.x & 7;
    const int rowBase = rb * ROWS_PER_BLOCK;
    const float* Xg   = X + ((size_t)b * NSEQ + rowBase) * DIM;

    const int t = threadIdx.x;
    stage_x_tile(Xg, Xs, t);
    stage_w_pairs(Wq, Wqp, t);
    stage_wait_and_barrier();

    const int wave = t >> 5;
    const int lane = t & 31;
    const int nl   = lane & 15;
    const int half = lane >> 4;
    const int col  = wave * 16 + nl;

    const float bqc = bq[col];
    const float kvc = kv[b * DIM + col];
    const float* Bq = &Wqp[half * WP_STRIDE + 2 * col];

    const v8f vzero = {};

    for (int mb = 0; mb < 2; ++mb) {
        v8f cq[4];
        const float* Arow[4];
        #pragma unroll
        for (int mm = 0; mm < 4; ++mm) {
            cq[mm] = vzero;
            Arow[mm] = &Xs[((mb * 4 + mm) * 16 + nl) * XS_STRIDE + 2 * half];
        }

        #pragma unroll 4
        for (int kk = 0; kk < DIM / 4; ++kk) {
            v2f bf = *(const v2f*)(Bq + (size_t)kk * 2 * WP_STRIDE);
            #pragma unroll
            for (int mm = 0; mm < 4; ++mm) {
                v2f a = *(const v2f*)(Arow[mm] + kk * 4);
                cq[mm] = __builtin_amdgcn_wmma_f32_16x16x4_f32(false, a, false, bf,
                                                               (short)0, cq[mm], false, false);
            }
        }

        // C/D layout: VGPR r, lane l -> (M = r + 8*half, N = nl)
        #pragma unroll
        for (int mm = 0; mm < 4; ++mm) {
            const size_t rowG0 = (size_t)b * NSEQ + rowBase + (mb * 4 + mm) * 16 + 8 * half;
            #pragma unroll
            for (int r = 0; r < 8; ++r)
                out[(rowG0 + r) * DIM + col] = (cq[mm][r] + bqc) * kvc;
        }
    }
}

// ---------------------------------------------------------------------------
extern "C" void kernel_launch(void* const* d_in, const int* in_sizes, int n_in,
                              void* d_out, int out_size, void* d_ws, size_t ws_size,
                              hipStream_t stream) {
    const float* X  = (const float*)d_in[0];
    const float* Wq = (const float*)d_in[1];
    const float* bq = (const float*)d_in[2];
    const float* Wk = (const float*)d_in[3];
    const float* bk = (const float*)d_in[4];
    const float* Wv = (const float*)d_in[5];
    const float* bv = (const float*)d_in[6];
    float* out = (float*)d_out;

    float* partial = (float*)d_ws;                                   // 32*128 floats
    float* kv      = partial + BATCH * ROWBLOCKS_PER_BATCH * DIM;    // 4*128 floats

    const int nBlocks = BATCH * ROWBLOCKS_PER_BATCH;                 // 32
    const size_t smem1 = ((size_t)ROWS_PER_BLOCK * XS_STRIDE +
                          2 * (size_t)(DIM / 2) * WP_STRIDE) * sizeof(float);
    const size_t smem3 = ((size_t)ROWS_PER_BLOCK * XS_STRIDE +
                          (size_t)(DIM / 2) * WP_STRIDE) * sizeof(float);

    kv_partial_kernel<<<nBlocks, 256, smem1, stream>>>(X, Wk, bk, Wv, bv, partial);
    kv_reduce_kernel<<<BATCH, DIM, 0, stream>>>(partial, kv);
    q_out_kernel<<<nBlocks, 256, smem3, stream>>>(X, Wq, bq, kv, out);
}